// UpdateTheta_67525475827838
// MI455X (gfx1250) — compile-verified
//
#include <hip/hip_runtime.h>

#define P 128
#define NB 64
#define LS 132          // padded LDS row stride: rows 16B-aligned, WMMA tiles conflict-free
#define HID 63          // (P-1)/2
#define PM1 127

typedef float v2f __attribute__((ext_vector_type(2)));
typedef float v8f __attribute__((ext_vector_type(8)));
typedef int   v4i __attribute__((ext_vector_type(4)));

// param block offsets
#define PB_GB1  0       // 63
#define PB_GW2  63      // 63
#define PB_GB2  126     // 1
#define PB_STW1 127     // 5
#define PB_STB1 132     // 5
#define PB_STW2 137     // 5
#define PB_STB2 142     // 1
#define PB_AW1  143     // 9
#define PB_AB1  152     // 3
#define PB_AW2  155     // 9
#define PB_AB2  164     // 3
#define PB_AW3  167     // 3
#define PB_AB3  170     // 1
#define PB_N    176

#if defined(__has_builtin)
#if __has_builtin(__builtin_amdgcn_global_load_async_to_lds_b128)
#define HAVE_ASYNC_LDS 1
#endif
#endif
#ifndef HAVE_ASYNC_LDS
#define HAVE_ASYNC_LDS 0
#endif

#if HAVE_ASYNC_LDS
#define GAS(p) ((__attribute__((address_space(1))) v4i*)(uintptr_t)(p))
#define LAS(p) ((__attribute__((address_space(3))) v4i*)(uintptr_t)(p))
__device__ __forceinline__ void wait_async0() {
#if __has_builtin(__builtin_amdgcn_s_wait_asynccnt)
    __builtin_amdgcn_s_wait_asynccnt(0);
#else
    asm volatile("s_wait_asynccnt 0" ::: "memory");
#endif
}
#endif

__global__ __launch_bounds__(256, 1)
void glasso_cd_kernel(const float* __restrict__ Theta_g,
                      const float* __restrict__ W_g,
                      const float* __restrict__ S_g,
                      const float* __restrict__ gw1_g, const float* __restrict__ gb1_g,
                      const float* __restrict__ gw2_g, const float* __restrict__ gb2_g,
                      const float* __restrict__ stw1_g, const float* __restrict__ stb1_g,
                      const float* __restrict__ stw2_g, const float* __restrict__ stb2_g,
                      const float* __restrict__ aw1_g, const float* __restrict__ ab1_g,
                      const float* __restrict__ aw2_g, const float* __restrict__ ab2_g,
                      const float* __restrict__ aw3_g, const float* __restrict__ ab3_g,
                      float* __restrict__ out)
{
    __shared__ __align__(16) float Wl[P * LS];       // W resident in LDS (symmetric)
    __shared__ __align__(16) float Tl[P * LS];       // Theta resident in LDS (symmetric)
    __shared__ __align__(16) float Sl[P * LS];       // S resident in LDS (symmetric)
    __shared__ __align__(16) float gw1[HID * PM1 + 1];
    __shared__ float prm[PB_N];
    __shared__ __align__(16) float u_s[P], z_s[P], t_s[P], y_s[P], v_s[P];
    __shared__ float part[256];
    __shared__ float redA[P], redB[P];
    __shared__ float scal[8];

    const int t = threadIdx.x;
    const int b = blockIdx.x;
    const float* Tg = Theta_g + (size_t)b * P * P;
    const float* Wg = W_g     + (size_t)b * P * P;
    const float* Sg = S_g     + (size_t)b * P * P;

    // ---- stage Theta, W, S, g_w1 into LDS ----
#if HAVE_ASYNC_LDS
    for (int i = 4 * t; i < P * P; i += 1024) {
        int r = i >> 7, c = i & 127;
        __builtin_amdgcn_global_load_async_to_lds_b128(GAS(Wg + i), LAS(&Wl[r * LS + c]), 0, 0);
        __builtin_amdgcn_global_load_async_to_lds_b128(GAS(Tg + i), LAS(&Tl[r * LS + c]), 0, 0);
        __builtin_amdgcn_global_load_async_to_lds_b128(GAS(Sg + i), LAS(&Sl[r * LS + c]), 0, 0);
    }
    for (int i = 4 * t; i < HID * PM1 - 3; i += 1024)
        __builtin_amdgcn_global_load_async_to_lds_b128(GAS(gw1_g + i), LAS(&gw1[i]), 0, 0);
    if (t == 0) gw1[HID * PM1 - 1] = gw1_g[HID * PM1 - 1];
    wait_async0();
#else
    for (int i = 4 * t; i < P * P; i += 1024) {
        int r = i >> 7, c = i & 127;
        *(float4*)&Wl[r * LS + c] = *(const float4*)(Wg + i);
        *(float4*)&Tl[r * LS + c] = *(const float4*)(Tg + i);
        *(float4*)&Sl[r * LS + c] = *(const float4*)(Sg + i);
    }
    for (int i = 4 * t; i < HID * PM1 - 3; i += 1024)
        *(float4*)&gw1[i] = *(const float4*)(gw1_g + i);
    if (t == 0) gw1[HID * PM1 - 1] = gw1_g[HID * PM1 - 1];
#endif
    if (t < HID) { prm[PB_GB1 + t] = gb1_g[t]; prm[PB_GW2 + t] = gw2_g[t]; }
    if (t == 0) {
        prm[PB_GB2] = gb2_g[0];
        for (int i = 0; i < 5; ++i) {
            prm[PB_STW1 + i] = stw1_g[i];
            prm[PB_STB1 + i] = stb1_g[i];
            prm[PB_STW2 + i] = stw2_g[i];
        }
        prm[PB_STB2] = stb2_g[0];
        for (int i = 0; i < 9; ++i) { prm[PB_AW1 + i] = aw1_g[i]; prm[PB_AW2 + i] = aw2_g[i]; }
        for (int i = 0; i < 3; ++i) { prm[PB_AB1 + i] = ab1_g[i]; prm[PB_AB2 + i] = ab2_g[i]; prm[PB_AW3 + i] = aw3_g[i]; }
        prm[PB_AB3] = ab3_g[0];
    }
    __syncthreads();

    for (int col = 0; col < P; ++col) {
        const float* Trow = &Tl[col * LS];      // Theta symmetric: column col == row col
        const float* Srow = &Sl[col * LS];      // S symmetric
        // ---- A: snapshot u = W[:,col] (== row col; W stays exactly symmetric) ----
        if (t < P) u_s[t] = Wl[col * LS + t];
        __syncthreads();
        const float w22  = u_s[col];
        const float s22  = Srow[col];
        const float th22 = Trow[col];

        // ---- B: gamma = |relu(theta12 @ gw1^T + gb1) @ gw2^T + gb2| ----
        {
            float p = 0.f;
            if (t < 4 * HID) {
                int k = t >> 2, q = t & 3;
                int j0 = q * 32;
                int j1 = (j0 + 32 < PM1) ? (j0 + 32) : PM1;
                const float* grow = &gw1[k * PM1];
                for (int j = j0; j < j1; ++j) {
                    int jm = j + (j >= col);    // skip index col
                    p += grow[j] * Trow[jm];
                }
                if (q == 0) p += prm[PB_GB1 + k];
            }
            part[t] = p;
        }
        __syncthreads();
        if (t < P) {
            float r = 0.f;
            if (t < HID) {
                float hid = part[4 * t] + part[4 * t + 1] + part[4 * t + 2] + part[4 * t + 3];
                r = fmaxf(hid, 0.f) * prm[PB_GW2 + t];
            }
            redA[t] = r;
        }
        __syncthreads();
        if (t < 32) {
            float s = redA[t] + redA[t + 32] + redA[t + 64] + redA[t + 96];
            for (int o = 16; o; o >>= 1) s += __shfl_xor(s, o);
            if (t == 0) scal[0] = fabsf(s + prm[PB_GB2]);
        }
        __syncthreads();
        const float gamma = scal[0];

        // ---- C: z (full vector, z[col]=0) ----
        if (t < P) z_s[t] = (t == col) ? 0.f : Trow[t] - gamma * (Srow[t] - u_s[t]);
        __syncthreads();

        // ---- D: y = W z ; quad = z^T W z - (u.z)^2/w22 ----
        {
            int i = t & 127, h = t >> 7;
            const float4* wr = (const float4*)&Wl[i * LS + h * 64];
            const float4* zz = (const float4*)&z_s[h * 64];
            float acc = 0.f;
            #pragma unroll
            for (int j = 0; j < 16; ++j) {
                float4 w = wr[j], z = zz[j];
                acc += w.x * z.x + w.y * z.y + w.z * z.z + w.w * z.w;
            }
            part[t] = acc;
        }
        __syncthreads();
        if (t < P) {
            float y = part[t] + part[t + 128];
            y_s[t] = y;
            redA[t] = u_s[t] * z_s[t];
            redB[t] = z_s[t] * y;
        }
        __syncthreads();
        if (t < 64) {
            float* rr = (t < 32) ? redA : redB;
            int l = t & 31;
            float s = rr[l] + rr[l + 32] + rr[l + 64] + rr[l + 96];
            for (int o = 16; o; o >>= 1) s += __shfl_xor(s, o);
            if (l == 0) scal[(t < 32) ? 2 : 3] = s;
        }
        __syncthreads();
        const float uz = scal[2], zWz = scal[3];
        const float quad = zWz - uz * uz / w22;
        const float rq = rsqrtf(quad);

        // ---- E: per-element shrinkage MLP + soft threshold ----
        if (t < P) {
            float zn = z_s[t] * rq;
            float lam = prm[PB_STB2];
            #pragma unroll
            for (int n = 0; n < 5; ++n)
                lam += fmaxf(zn * prm[PB_STW1 + n] + prm[PB_STB1 + n], 0.f) * prm[PB_STW2 + n];
            lam = fabsf(lam);
            t_s[t] = (t == col) ? 0.f : copysignf(fmaxf(fabsf(zn) - lam, 0.f), zn);
        }
        __syncthreads();

        // ---- F: y2 = W t ; quad2 = t^T W t - (u.t)^2/w22 ----
        {
            int i = t & 127, h = t >> 7;
            const float4* wr = (const float4*)&Wl[i * LS + h * 64];
            const float4* zz = (const float4*)&t_s[h * 64];
            float acc = 0.f;
            #pragma unroll
            for (int j = 0; j < 16; ++j) {
                float4 w = wr[j], z = zz[j];
                acc += w.x * z.x + w.y * z.y + w.z * z.z + w.w * z.w;
            }
            part[t] = acc;
        }
        __syncthreads();
        if (t < P) {
            float y = part[t] + part[t + 128];
            y_s[t] = y;
            redA[t] = u_s[t] * t_s[t];
            redB[t] = t_s[t] * y;
        }
        __syncthreads();
        if (t < 64) {
            float* rr = (t < 32) ? redA : redB;
            int l = t & 31;
            float s = rr[l] + rr[l + 32] + rr[l + 64] + rr[l + 96];
            for (int o = 16; o; o >>= 1) s += __shfl_xor(s, o);
            if (l == 0) scal[(t < 32) ? 2 : 3] = s;
        }
        __syncthreads();
        const float ut = scal[2], tWt = scal[3];
        const float quad2 = tWt - ut * ut / w22;

        // ---- G: alpha via 3->3->3->1 MLP (single thread, broadcast) ----
        if (t == 0) {
            float f0 = quad2, f1 = s22, f2 = th22;
            float h1[3], h2[3];
            #pragma unroll
            for (int m = 0; m < 3; ++m)
                h1[m] = fmaxf(prm[PB_AW1 + 3 * m] * f0 + prm[PB_AW1 + 3 * m + 1] * f1 +
                              prm[PB_AW1 + 3 * m + 2] * f2 + prm[PB_AB1 + m], 0.f);
            #pragma unroll
            for (int m = 0; m < 3; ++m)
                h2[m] = fmaxf(prm[PB_AW2 + 3 * m] * h1[0] + prm[PB_AW2 + 3 * m + 1] * h1[1] +
                              prm[PB_AW2 + 3 * m + 2] * h1[2] + prm[PB_AB2 + m], 0.f);
            scal[5] = fabsf(prm[PB_AW3] * h2[0] + prm[PB_AW3 + 1] * h2[1] +
                            prm[PB_AW3 + 2] * h2[2] + prm[PB_AB3]);
        }
        __syncthreads();
        const float alpha = scal[5];
        const float t22n  = alpha + quad2;
        const float w22n  = 1.f / alpha;        // 1/(t22n - quad2)

        // ---- H/I: build v (w12n extended with w22n) and update Theta row/col/diag ----
        if (t < P) {
            v_s[t] = (t == col) ? w22n : -w22n * (y_s[t] - u_s[t] * ut / w22);
            if (t == col) {
                Tl[col * LS + col] = t22n;
            } else {
                float tv = t_s[t];
                Tl[t * LS + col] = tv;
                Tl[col * LS + t] = tv;
            }
        }
        __syncthreads();

        // ---- J: rank-2 symmetric update  W += (-u u^T)/w22 + (v v^T)/w22n
        //      via V_WMMA_F32_16X16X4_F32: A=[-u/w22 | v*alpha | 0 | 0] (16x4),
        //      B=[u^T ; v^T ; 0 ; 0] (4x16), C/D = 16x16 W tile (8 tiles/wave).
        {
            const float su = -1.f / w22;   // scale of u-column in A
            const float sv = alpha;        // 1/w22n
            const int wv = t >> 5;         // wave id 0..7 -> row-tile I
            const int lane = t & 31;
            float a0 = 0.f, a1 = 0.f;
            if (lane < 16) {
                a0 = u_s[wv * 16 + lane] * su;
                a1 = v_s[wv * 16 + lane] * sv;
            }
            v2f A = {a0, a1};              // lanes 0-15: K=0,1 ; lanes 16-31: K=2,3 (=0)
            const int rrow = wv * 16 + ((lane >> 4) << 3);   // M base for C/D regs
            const int cb = lane & 15;
            for (int J = 0; J < 8; ++J) {
                float b0 = 0.f, b1 = 0.f;
                if (lane < 16) { b0 = u_s[J * 16 + lane]; b1 = v_s[J * 16 + lane]; }
                v2f Bv = {b0, b1};
                v8f C;
                const int base = rrow * LS + J * 16 + cb;
                #pragma unroll
                for (int r = 0; r < 8; ++r) C[r] = Wl[base + r * LS];
                v8f D = __builtin_amdgcn_wmma_f32_16x16x4_f32(
                            false, A, false, Bv, (short)0, C, false, false);
                #pragma unroll
                for (int r = 0; r < 8; ++r) Wl[base + r * LS] = D[r];
            }
        }
        __syncthreads();
    }

    // ---- write Theta back (vectorized) ----
    for (int i = 4 * t; i < P * P; i += 1024) {
        int r = i >> 7, c = i & 127;
        *(float4*)(out + (size_t)b * P * P + i) = *(const float4*)&Tl[r * LS + c];
    }
}

extern "C" void kernel_launch(void* const* d_in, const int* in_sizes, int n_in,
                              void* d_out, int out_size, void* d_ws, size_t ws_size,
                              hipStream_t stream) {
    (void)in_sizes; (void)n_in; (void)out_size; (void)d_ws; (void)ws_size;
    const float* Theta = (const float*)d_in[0];
    const float* W     = (const float*)d_in[1];
    const float* S     = (const float*)d_in[2];
    const float* gw1   = (const float*)d_in[3];
    const float* gb1   = (const float*)d_in[4];
    const float* gw2   = (const float*)d_in[5];
    const float* gb2   = (const float*)d_in[6];
    const float* stw1  = (const float*)d_in[7];
    const float* stb1  = (const float*)d_in[8];
    const float* stw2  = (const float*)d_in[9];
    const float* stb2  = (const float*)d_in[10];
    const float* aw1   = (const float*)d_in[11];
    const float* ab1   = (const float*)d_in[12];
    const float* aw2   = (const float*)d_in[13];
    const float* ab2   = (const float*)d_in[14];
    const float* aw3   = (const float*)d_in[15];
    const float* ab3   = (const float*)d_in[16];
    float* out = (float*)d_out;

    glasso_cd_kernel<<<dim3(NB), dim3(256), 0, stream>>>(
        Theta, W, S, gw1, gb1, gw2, gb2, stw1, stb1, stw2, stb2,
        aw1, ab1, aw2, ab2, aw3, ab3, out);
}